// ACNN_79946521247827
// MI455X (gfx1250) — compile-verified
//
#include <hip/hip_runtime.h>
#include <hip/hip_bf16.h>

// ---------------------------------------------------------------------------
// Diffractive-optics CNN forward for MI455X (gfx1250).
// 167 is prime -> FFTs are dense DFT matmuls on V_WMMA_F32_16X16X4_F32.
// All planes padded to 176x176 so the GEMM inner loop has ZERO bounds checks:
// pure b64/b32 loads + 4 WMMAs per K-step.  One wave32 per 16x16 C tile.
// ---------------------------------------------------------------------------

typedef float v2f __attribute__((ext_vector_type(2)));
typedef float v8f __attribute__((ext_vector_type(8)));

#define NN   167
#define PN   176                 // padded edge (11 * 16)
#define PN2  (PN * PN)           // padded plane stride (30976 floats)
#define TIL  11
#define PI_F 3.14159265358979323846f

__device__ __forceinline__ v8f wmma4(v2f a, v2f b, v8f c) {
  // D = A(16x4) * B(4x16) + C, fp32 — CDNA5 V_WMMA_F32_16X16X4_F32
  return __builtin_amdgcn_wmma_f32_16x16x4_f32(
      /*neg_a=*/false, a, /*neg_b=*/false, b,
      /*c_mod=*/(short)0, c, /*reuse_a=*/false, /*reuse_b=*/false);
}

// ---------------- constant builders -----------------------------------------

__global__ void build_dft(float* __restrict__ Wfr, float* __restrict__ Wfi,
                          float* __restrict__ WIr, float* __restrict__ WIi) {
  int idx = blockIdx.x * blockDim.x + threadIdx.x;
  if (idx >= PN2) return;
  int j = idx / PN, k = idx % PN;
  float cr = 0.f, ci = 0.f;
  if (j < NN && k < NN) {
    int m = (j * k) % NN;                     // exact phase reduction
    float ang = -2.0f * PI_F * (float)m / (float)NN;
    float s, c;
    __sincosf(ang, &s, &c);
    cr = c; ci = s;
  }
  Wfr[idx] = cr;  Wfi[idx] = ci;              // forward DFT (pad = 0)
  WIr[idx] = cr;  WIi[idx] = -ci;             // inverse (conj); 1/N via gemm scale
}

__global__ void build_amp(float* __restrict__ ampInv) {
  if (threadIdx.x == 0 && blockIdx.x == 0) {
    int cnt = 0;
    for (int i = 0; i < 84; ++i)
      for (int j = 0; j < 84; ++j) {
        float y = (float)i - 41.5f, x = (float)j - 41.5f;
        if (x * x + y * y <= 1764.0f) ++cnt;   // R = 42
      }
    ampInv[0] = 1.0f / (float)cnt;
  }
}

// ---------------- complex GEMMs via f32 WMMA --------------------------------
// Padded 176x176 operands: the inner loop is branch-free.

// C[b] = W (shared, padded) @ X[b]
__global__ void cgemm_left(const float* __restrict__ Ar, const float* __restrict__ Ai,
                           const float* __restrict__ Xr, const float* __restrict__ Xi,
                           float* __restrict__ Cr, float* __restrict__ Ci, float scale) {
  const int lane = threadIdx.x;
  const int hlf = lane >> 4, r = lane & 15;
  const int tm = (blockIdx.x / TIL) * 16, tn = (blockIdx.x % TIL) * 16;
  const size_t b = blockIdx.y;
  const float* xr = Xr + b * PN2;
  const float* xi = Xi + b * PN2;
  v8f rr = {}, ii = {}, ri = {}, ir = {};
  const int m = tm + r, n = tn + r;
  const float* arp = Ar + m * PN;           // A row, contiguous K
  const float* aip = Ai + m * PN;
  #pragma unroll 4
  for (int kb = 0; kb < PN; kb += 4) {
    const int k0 = kb + 2 * hlf;
    v2f ar = *(const v2f*)(arp + k0);       // aligned b64 load
    v2f ai = *(const v2f*)(aip + k0);
    v2f br, bi;
    br.x = xr[k0 * PN + n];       br.y = xr[(k0 + 1) * PN + n];
    bi.x = xi[k0 * PN + n];       bi.y = xi[(k0 + 1) * PN + n];
    rr = wmma4(ar, br, rr);
    ii = wmma4(ai, bi, ii);
    ri = wmma4(ar, bi, ri);
    ir = wmma4(ai, br, ir);
  }
  float* cr = Cr + b * PN2;
  float* ci = Ci + b * PN2;
  #pragma unroll
  for (int v = 0; v < 8; ++v) {
    const int mo = tm + v + 8 * hlf, no = tn + r;
    cr[mo * PN + no] = (rr[v] - ii[v]) * scale;
    ci[mo * PN + no] = (ri[v] + ir[v]) * scale;
  }
}

// C[b] = X[b] @ W (shared, padded)
__global__ void cgemm_right(const float* __restrict__ Xr, const float* __restrict__ Xi,
                            const float* __restrict__ Br, const float* __restrict__ Bi,
                            float* __restrict__ Cr, float* __restrict__ Ci, float scale) {
  const int lane = threadIdx.x;
  const int hlf = lane >> 4, r = lane & 15;
  const int tm = (blockIdx.x / TIL) * 16, tn = (blockIdx.x % TIL) * 16;
  const size_t b = blockIdx.y;
  const float* xr = Xr + b * PN2;
  const float* xi = Xi + b * PN2;
  v8f rr = {}, ii = {}, ri = {}, ir = {};
  const int m = tm + r, n = tn + r;
  const float* arp = xr + m * PN;
  const float* aip = xi + m * PN;
  #pragma unroll 4
  for (int kb = 0; kb < PN; kb += 4) {
    const int k0 = kb + 2 * hlf;
    v2f ar = *(const v2f*)(arp + k0);
    v2f ai = *(const v2f*)(aip + k0);
    v2f br, bi;
    br.x = Br[k0 * PN + n];       br.y = Br[(k0 + 1) * PN + n];
    bi.x = Bi[k0 * PN + n];       bi.y = Bi[(k0 + 1) * PN + n];
    rr = wmma4(ar, br, rr);
    ii = wmma4(ai, bi, ii);
    ri = wmma4(ar, bi, ri);
    ir = wmma4(ai, br, ir);
  }
  float* cr = Cr + b * PN2;
  float* ci = Ci + b * PN2;
  #pragma unroll
  for (int v = 0; v < 8; ++v) {
    const int mo = tm + v + 8 * hlf, no = tn + r;
    cr[mo * PN + no] = (rr[v] - ii[v]) * scale;
    ci[mo * PN + no] = (ri[v] + ir[v]) * scale;
  }
}

// ---------------- pointwise stages (write zero into the 176-pad apron) ------

// pupil: Pc = mask * exp(iP), padded (42,41) into 167, then into 176 apron
__global__ void pupil_pad(const float* __restrict__ P,
                          float* __restrict__ Pr, float* __restrict__ Pi) {
  int idx = blockIdx.x * blockDim.x + threadIdx.x;
  if (idx >= 16 * PN2) return;
  int t = idx % PN2, u = t / PN, v = t % PN, b = idx / PN2;
  float re = 0.f, im = 0.f;
  int i = u - 42, j = v - 42;
  if (u < NN && v < NN && i >= 0 && i < 84 && j >= 0 && j < 84) {
    float yy = (float)i - 41.5f, xx = (float)j - 41.5f;
    if (xx * xx + yy * yy <= 1764.0f) {
      float s, c;
      __sincosf(P[(b * 84 + i) * 84 + j], &s, &c);
      re = c; im = s;
    }
  }
  Pr[idx] = re; Pi[idx] = im;
}

__global__ void mag2_inplace(float* __restrict__ Fr, float* __restrict__ Fi) {
  int idx = blockIdx.x * blockDim.x + threadIdx.x;
  if (idx >= 16 * PN2) return;
  float a = Fr[idx], b = Fi[idx];
  Fr[idx] = a * a + b * b;   // F * conj(F)  (pad stays 0)
  Fi[idx] = 0.f;
}

// pad 90->167 (39,38) with pre-FFT fftshift folded into the gather
__global__ void pad_shift_x(const float* __restrict__ x,
                            float* __restrict__ Xr, float* __restrict__ Xi) {
  int idx = blockIdx.x * blockDim.x + threadIdx.x;
  if (idx >= 16 * PN2) return;
  int t = idx % PN2, u = t / PN, v = t % PN, b = idx / PN2;
  float re = 0.f;
  if (u < NN && v < NN) {
    int su = (u + 84) % NN, sv = (v + 84) % NN;  // fftshift (odd N, shift 83)
    int i = su - 39, j = sv - 39;
    if (i >= 0 && i < 90 && j >= 0 && j < 90)
      re = x[(b * 90 + i) * 90 + j];
  }
  Xr[idx] = re; Xi[idx] = 0.f;
}

// ys[b] = shifted( X[b/16] * OTF[b%16] ); two odd-N shifts of 84 compose
// to a single +1 circular shift.
__global__ void cross_mul(const float* __restrict__ X0r, const float* __restrict__ X0i,
                          const float* __restrict__ Zr,  const float* __restrict__ Zi,
                          const float* __restrict__ ampInv,
                          float* __restrict__ Yr, float* __restrict__ Yi) {
  int idx = blockIdx.x * blockDim.x + threadIdx.x;
  if (idx >= 256 * PN2) return;
  int t = idx % PN2, u = t / PN, v = t % PN, b = idx / PN2;
  float yr = 0.f, yi = 0.f;
  if (u < NN && v < NN) {
    int su = (u + 1) % NN, sv = (v + 1) % NN;
    int s = su * PN + sv;
    int img = b >> 4, ker = b & 15;
    float ainv = ampInv[0];
    float ar = X0r[img * PN2 + s], ai = X0i[img * PN2 + s];
    float br = Zr[ker * PN2 + s] * ainv, bi = Zi[ker * PN2 + s] * ainv;
    yr = ar * br - ai * bi;
    yi = ar * bi + ai * br;
  }
  Yr[idx] = yr; Yi[idx] = yi;
}

// post-ifft fftshift + kernel diff (0..7 minus 8..15) + noise + crop(24:144)
// + relu + 3x3/3 maxpool, fused.  pooled layout matches reshape(16,-1).
__global__ void diff_pool(const float* __restrict__ Zre, float* __restrict__ pooled) {
  int q = blockIdx.x * blockDim.x + threadIdx.x;
  if (q >= 16 * 12800) return;
  int c = q & 7, t = q >> 3;
  int px = t % 40; t /= 40;
  int py = t % 40; int img = t / 40;
  const float* a = Zre + (size_t)(img * 16 + c) * PN2;
  const float* b = Zre + (size_t)(img * 16 + c + 8) * PN2;
  float mx = 0.0f;                               // relu floor
  for (int wy = 0; wy < 3; ++wy)
    for (int wx = 0; wx < 3; ++wx) {
      int u = 24 + 3 * py + wy, v = 24 + 3 * px + wx;
      int su = (u + 84) % NN, sv = (v + 84) % NN; // post-ifft fftshift
      int s = su * PN + sv;
      // fixed pseudo-noise standing in for the key(42) constant tensor
      unsigned h = (unsigned)(((img * NN + u) * NN + v) * 8 + c) * 2654435761u;
      float nz = 0.003f * (((h >> 9) * (1.0f / 4194304.0f)) - 1.0f);
      mx = fmaxf(mx, a[s] - b[s] + nz);
    }
  pooled[q] = mx;
}

// logits[16x16] = pooled[16,12800] @ W3[12800,10]  (single wave, f32 WMMA)
__global__ void dense_wmma(const float* __restrict__ flat, const float* __restrict__ W3,
                           float* __restrict__ logits) {
  const int lane = threadIdx.x;
  const int hlf = lane >> 4, r = lane & 15;
  v8f acc = {};
  for (int kb = 0; kb < 12800; kb += 4) {
    const int k0 = kb + 2 * hlf;
    v2f a, b;
    a.x = flat[r * 12800 + k0];
    a.y = flat[r * 12800 + k0 + 1];
    b.x = (r < 10) ? W3[k0 * 10 + r] : 0.f;
    b.y = (r < 10) ? W3[(k0 + 1) * 10 + r] : 0.f;
    acc = wmma4(a, b, acc);
  }
  #pragma unroll
  for (int v = 0; v < 8; ++v)
    logits[(v + 8 * hlf) * 16 + r] = acc[v];
}

__global__ void softmax16(const float* __restrict__ logits, const float* __restrict__ b3,
                          float* __restrict__ out) {
  int i = threadIdx.x;
  if (i >= 16) return;
  float v[10], mx = -1e30f;
  for (int j = 0; j < 10; ++j) { v[j] = logits[i * 16 + j] + b3[j]; mx = fmaxf(mx, v[j]); }
  float s = 0.f;
  for (int j = 0; j < 10; ++j) { v[j] = __expf(v[j] - mx); s += v[j]; }
  float inv = 1.0f / s;
  for (int j = 0; j < 10; ++j) out[i * 10 + j] = v[j] * inv;
}

// ---------------- driver -----------------------------------------------------

extern "C" void kernel_launch(void* const* d_in, const int* in_sizes, int n_in,
                              void* d_out, int out_size, void* d_ws, size_t ws_size,
                              hipStream_t stream) {
  const float* x  = (const float*)d_in[0];   // [16,90,90,1]
  const float* P  = (const float*)d_in[1];   // [16,84,84]
  const float* W3 = (const float*)d_in[2];   // [12800,10]
  const float* b3 = (const float*)d_in[3];   // [10]
  float* out = (float*)d_out;                // [16,10]

  float* ws = (float*)d_ws;
  size_t o = 0;
  float* Wfr = ws + o; o += PN2;
  float* Wfi = ws + o; o += PN2;
  float* WIr = ws + o; o += PN2;
  float* WIi = ws + o; o += PN2;
  float* amp = ws + o; o += 16;
  const size_t S16 = (size_t)16 * PN2;
  float* B0r = ws + o; o += S16;  float* B0i = ws + o; o += S16;
  float* B1r = ws + o; o += S16;  float* B1i = ws + o; o += S16;
  float* B2r = ws + o; o += S16;  float* B2i = ws + o; o += S16;
  const size_t S256 = (size_t)256 * PN2;
  float* Yr  = ws + o; o += S256; float* Yi  = ws + o; o += S256;
  float* Y2r = ws + o; o += S256; float* Y2i = ws + o; o += S256;
  float* pooled = ws + o; o += (size_t)16 * 12800;
  float* logit  = ws + o; o += 256;

  const dim3 wv(32);
  const dim3 g16(TIL * TIL, 16), g256(TIL * TIL, 256);
  const float invN = 1.0f / (float)NN;

  // constants
  build_dft<<<(PN2 + 255) / 256, 256, 0, stream>>>(Wfr, Wfi, WIr, WIi);
  build_amp<<<1, 1, 0, stream>>>(amp);

  // OTF: fft2(pupil) -> |F|^2 -> ifft2  (autocorrelation theorem)
  pupil_pad<<<(16 * PN2 + 255) / 256, 256, 0, stream>>>(P, B0r, B0i);
  cgemm_left <<<g16, wv, 0, stream>>>(Wfr, Wfi, B0r, B0i, B1r, B1i, 1.0f);
  cgemm_right<<<g16, wv, 0, stream>>>(B1r, B1i, Wfr, Wfi, B0r, B0i, 1.0f);
  mag2_inplace<<<(16 * PN2 + 255) / 256, 256, 0, stream>>>(B0r, B0i);
  cgemm_left <<<g16, wv, 0, stream>>>(WIr, WIi, B0r, B0i, B1r, B1i, invN);
  cgemm_right<<<g16, wv, 0, stream>>>(B1r, B1i, WIr, WIi, B2r, B2i, invN);  // Z (pre-shift OTF)

  // input field: pad+shift, forward fft2
  pad_shift_x<<<(16 * PN2 + 255) / 256, 256, 0, stream>>>(x, B0r, B0i);
  cgemm_left <<<g16, wv, 0, stream>>>(Wfr, Wfi, B0r, B0i, B1r, B1i, 1.0f);
  cgemm_right<<<g16, wv, 0, stream>>>(B1r, B1i, Wfr, Wfi, B0r, B0i, 1.0f);  // X0

  // batch-256 frequency product, inverse fft2
  cross_mul<<<(256 * PN2 + 255) / 256, 256, 0, stream>>>(B0r, B0i, B2r, B2i, amp, Yr, Yi);
  cgemm_left <<<g256, wv, 0, stream>>>(WIr, WIi, Yr, Yi, Y2r, Y2i, invN);
  cgemm_right<<<g256, wv, 0, stream>>>(Y2r, Y2i, WIr, WIi, Yr, Yi, invN);   // final field

  // head
  diff_pool<<<(16 * 12800 + 255) / 256, 256, 0, stream>>>(Yr, pooled);
  dense_wmma<<<1, wv, 0, stream>>>(pooled, W3, logit);
  softmax16<<<1, wv, 0, stream>>>(logit, b3, out);
}